// TGNEncoder_74234214744807
// MI455X (gfx1250) — compile-verified
//
#include <hip/hip_runtime.h>
#include <math.h>

#define N_NODES 50000
#define N_EDGES 800000
#define IN_DIM  128
#define MEM_DIM 32
#define Z_DIM   160
#define OUT_DIM 64

typedef float v2f __attribute__((ext_vector_type(2)));
typedef float v8f __attribute__((ext_vector_type(8)));

// ---- ordered-uint transform for f32 atomic max (handles negatives) ----
__device__ __forceinline__ unsigned f2ord(float f) {
  unsigned u = __float_as_uint(f);
  return (u & 0x80000000u) ? ~u : (u | 0x80000000u);
}
__device__ __forceinline__ float ord2f(unsigned u) {
  return (u & 0x80000000u) ? __uint_as_float(u & 0x7FFFFFFFu)
                           : __uint_as_float(~u);
}

__global__ void init_stats_kernel(unsigned* __restrict__ smax_u, float* __restrict__ ssum) {
  int i = blockIdx.x * blockDim.x + threadIdx.x;
  if (i < N_NODES * 2) { smax_u[i] = 0u; ssum[i] = 0.0f; }
}

// ---- Pass 1: [16-node tile] x [160] @ [160 x 64] with V_WMMA_F32_16X16X4_F32
// blockIdx.y selects which weight matrix / output buffer.
__global__ __launch_bounds__(128)
void node_proj_kernel(const float* __restrict__ x, const float* __restrict__ mem,
                      const float* __restrict__ Wq, const float* __restrict__ Wk,
                      const float* __restrict__ Wv, const float* __restrict__ Wskip,
                      float* __restrict__ zq, float* __restrict__ zk,
                      float* __restrict__ zv, float* __restrict__ outskip) {
  const int lane = threadIdx.x & 31;
  const int wid  = threadIdx.x >> 5;
  const int tile = blockIdx.x * 4 + wid;
  if (tile >= N_NODES / 16) return;

  const float* W; float* ob;
  switch (blockIdx.y) {
    case 0:  W = Wq;    ob = zq;      break;
    case 1:  W = Wk;    ob = zk;      break;
    case 2:  W = Wv;    ob = zv;      break;
    default: W = Wskip; ob = outskip; break;
  }

  const int base = tile * 16;
  const int half = lane >> 4;        // 0: K{0,1}, 1: K{2,3}
  const int lx   = lane & 15;
  const int row  = base + lx;        // A-matrix row owned by this lane
  const int koff = half * 2;

  const v8f vzero = {0.f,0.f,0.f,0.f,0.f,0.f,0.f,0.f};
  v8f acc[4] = {vzero, vzero, vzero, vzero};

  for (int kt = 0; kt < Z_DIM / 4; ++kt) {
    const int c0 = kt * 4 + koff;
    v2f a;
    if (c0 < IN_DIM) {               // z = [x || memory]; 128 % 4 == 0 so pairs never straddle
      const float* p = x + (size_t)row * IN_DIM + c0;
      a.x = p[0]; a.y = p[1];
    } else {
      const float* p = mem + (size_t)row * MEM_DIM + (c0 - IN_DIM);
      a.x = p[0]; a.y = p[1];
    }
#pragma unroll
    for (int nt = 0; nt < 4; ++nt) {
      const int col = nt * 16 + lx;
      v2f b;
      b.x = W[(size_t)c0       * OUT_DIM + col];
      b.y = W[(size_t)(c0 + 1) * OUT_DIM + col];
      acc[nt] = __builtin_amdgcn_wmma_f32_16x16x4_f32(false, a, false, b,
                                                      (short)0, acc[nt], false, false);
    }
  }
#pragma unroll
  for (int nt = 0; nt < 4; ++nt) {
    const int col = nt * 16 + lx;
#pragma unroll
    for (int i = 0; i < 8; ++i) {
      const int r = base + i + half * 8;
      ob[(size_t)r * OUT_DIM + col] = acc[nt][i];
    }
  }
}

// ---- Pass 2: te = cos(t*w+b) per 16-edge tile, tek = te@We_k via WMMA,
// k = tek + zk[src], score = <q,k>/sqrt(32) via 16-lane shuffles, atomicMax stats.
__global__ __launch_bounds__(128)
void edge_score_kernel(const int* __restrict__ src, const int* __restrict__ dst,
                       const float* __restrict__ t,
                       const float* __restrict__ w_time, const float* __restrict__ b_time,
                       const float* __restrict__ Wek,
                       const float* __restrict__ zq, const float* __restrict__ zk,
                       float* __restrict__ scoreArr, unsigned* __restrict__ smax_u) {
  const int lane = threadIdx.x & 31;
  const int wid  = threadIdx.x >> 5;
  const int tile = blockIdx.x * 4 + wid;
  if (tile >= N_EDGES / 16) return;
  const int base = tile * 16;
  const int half = lane >> 4;
  const int lx   = lane & 15;
  const int koff = half * 2;
  const float tv = t[base + lx];

  const v8f vzero = {0.f,0.f,0.f,0.f,0.f,0.f,0.f,0.f};
  v8f acc[4] = {vzero, vzero, vzero, vzero};

#pragma unroll
  for (int kt = 0; kt < MEM_DIM / 4; ++kt) {
    const int c0 = kt * 4 + koff;
    v2f a;
    a.x = __cosf(tv * w_time[c0]     + b_time[c0]);
    a.y = __cosf(tv * w_time[c0 + 1] + b_time[c0 + 1]);
#pragma unroll
    for (int nt = 0; nt < 4; ++nt) {
      const int col = nt * 16 + lx;
      v2f b;
      b.x = Wek[(size_t)c0       * OUT_DIM + col];
      b.y = Wek[(size_t)(c0 + 1) * OUT_DIM + col];
      acc[nt] = __builtin_amdgcn_wmma_f32_16x16x4_f32(false, a, false, b,
                                                      (short)0, acc[nt], false, false);
    }
  }

  const float scale = 0.17677669529663687f;  // 1/sqrt(32)
#pragma unroll
  for (int i = 0; i < 8; ++i) {
    const int row = i + half * 8;
    const int e   = base + row;
    const int s   = src[e];
    const int d   = dst[e];
    float sc0 = 0.f, sc1 = 0.f;
#pragma unroll
    for (int nt = 0; nt < 4; ++nt) {
      const int col = nt * 16 + lx;
      float kf   = acc[nt][i] + zk[(size_t)s * OUT_DIM + col];
      float qv   = zq[(size_t)d * OUT_DIM + col];
      float prod = kf * qv;
      // reduce across the 16 lanes of this half-wave
      for (int off = 1; off < 16; off <<= 1) prod += __shfl_xor(prod, off, 16);
      if ((nt >> 1) == 0) sc0 += prod; else sc1 += prod;
    }
    if (lx == 0) {
      float v0 = sc0 * scale, v1 = sc1 * scale;
      scoreArr[(size_t)e * 2 + 0] = v0;
      scoreArr[(size_t)e * 2 + 1] = v1;
      atomicMax(&smax_u[(size_t)d * 2 + 0], f2ord(v0));
      atomicMax(&smax_u[(size_t)d * 2 + 1], f2ord(v1));
    }
  }
}

// ---- Pass 3: p = exp(score - smax[dst]); ssum[dst] += p (in-place on scoreArr)
__global__ void softmax_kernel(const int* __restrict__ dst, float* __restrict__ scoreArr,
                               const unsigned* __restrict__ smax_u, float* __restrict__ ssum) {
  int e = blockIdx.x * blockDim.x + threadIdx.x;
  if (e >= N_EDGES) return;
  int d = dst[e];
#pragma unroll
  for (int h = 0; h < 2; ++h) {
    float m = ord2f(smax_u[(size_t)d * 2 + h]);
    float p = __expf(scoreArr[(size_t)e * 2 + h] - m);
    scoreArr[(size_t)e * 2 + h] = p;
    atomicAdd(&ssum[(size_t)d * 2 + h], p);
  }
}

// ---- Pass 4: recompute tev = te@We_v via WMMA, v = tev + zv[src],
// out[dst] += alpha * v (alpha = p / ssum[dst]) with f32 atomics.
__global__ __launch_bounds__(128)
void edge_agg_kernel(const int* __restrict__ src, const int* __restrict__ dst,
                     const float* __restrict__ t,
                     const float* __restrict__ w_time, const float* __restrict__ b_time,
                     const float* __restrict__ Wev,
                     const float* __restrict__ zv,
                     const float* __restrict__ pArr, const float* __restrict__ ssum,
                     float* __restrict__ out) {
  const int lane = threadIdx.x & 31;
  const int wid  = threadIdx.x >> 5;
  const int tile = blockIdx.x * 4 + wid;
  if (tile >= N_EDGES / 16) return;
  const int base = tile * 16;
  const int half = lane >> 4;
  const int lx   = lane & 15;
  const int koff = half * 2;
  const float tv = t[base + lx];

  const v8f vzero = {0.f,0.f,0.f,0.f,0.f,0.f,0.f,0.f};
  v8f acc[4] = {vzero, vzero, vzero, vzero};

#pragma unroll
  for (int kt = 0; kt < MEM_DIM / 4; ++kt) {
    const int c0 = kt * 4 + koff;
    v2f a;
    a.x = __cosf(tv * w_time[c0]     + b_time[c0]);
    a.y = __cosf(tv * w_time[c0 + 1] + b_time[c0 + 1]);
#pragma unroll
    for (int nt = 0; nt < 4; ++nt) {
      const int col = nt * 16 + lx;
      v2f b;
      b.x = Wev[(size_t)c0       * OUT_DIM + col];
      b.y = Wev[(size_t)(c0 + 1) * OUT_DIM + col];
      acc[nt] = __builtin_amdgcn_wmma_f32_16x16x4_f32(false, a, false, b,
                                                      (short)0, acc[nt], false, false);
    }
  }

#pragma unroll
  for (int i = 0; i < 8; ++i) {
    const int row = i + half * 8;
    const int e   = base + row;
    const int s   = src[e];
    const int d   = dst[e];
    float a0 = pArr[(size_t)e * 2 + 0] / ssum[(size_t)d * 2 + 0];
    float a1 = pArr[(size_t)e * 2 + 1] / ssum[(size_t)d * 2 + 1];
#pragma unroll
    for (int nt = 0; nt < 4; ++nt) {
      const int col = nt * 16 + lx;
      float v = acc[nt][i] + zv[(size_t)s * OUT_DIM + col];
      float alpha = (nt >> 1) ? a1 : a0;
      atomicAdd(&out[(size_t)d * OUT_DIM + col], alpha * v);
    }
  }
}

extern "C" void kernel_launch(void* const* d_in, const int* in_sizes, int n_in,
                              void* d_out, int out_size, void* d_ws, size_t ws_size,
                              hipStream_t stream) {
  (void)in_sizes; (void)n_in; (void)out_size; (void)ws_size;
  const int*   src    = (const int*)  d_in[0];
  const int*   dst    = (const int*)  d_in[1];
  const float* t      = (const float*)d_in[2];
  const float* x      = (const float*)d_in[3];
  const float* mem    = (const float*)d_in[4];
  const float* w_time = (const float*)d_in[5];
  const float* b_time = (const float*)d_in[6];
  const float* Wq     = (const float*)d_in[7];
  const float* Wk     = (const float*)d_in[8];
  const float* Wv     = (const float*)d_in[9];
  const float* Wek    = (const float*)d_in[10];
  const float* Wev    = (const float*)d_in[11];
  const float* Wskip  = (const float*)d_in[12];
  float* out = (float*)d_out;

  // workspace layout (~46 MB)
  float* ws = (float*)d_ws;
  float*    zq       = ws;
  float*    zk       = zq + (size_t)N_NODES * OUT_DIM;
  float*    zv       = zk + (size_t)N_NODES * OUT_DIM;
  float*    scoreArr = zv + (size_t)N_NODES * OUT_DIM;          // E*2
  unsigned* smax_u   = (unsigned*)(scoreArr + (size_t)N_EDGES * 2);
  float*    ssum     = (float*)(smax_u + (size_t)N_NODES * 2);

  // Pass 0: reset softmax stats
  init_stats_kernel<<<(N_NODES * 2 + 255) / 256, 256, 0, stream>>>(smax_u, ssum);

  // Pass 1: node projections (y=3 writes skip directly into d_out)
  dim3 g1((N_NODES / 16 + 3) / 4, 4);
  node_proj_kernel<<<g1, 128, 0, stream>>>(x, mem, Wq, Wk, Wv, Wskip, zq, zk, zv, out);

  // Pass 2: edge scores + segment max (50000 tiles, 4 waves/block)
  edge_score_kernel<<<(N_EDGES / 16) / 4, 128, 0, stream>>>(
      src, dst, t, w_time, b_time, Wek, zq, zk, scoreArr, smax_u);

  // Pass 3: exp + segment sum
  softmax_kernel<<<(N_EDGES + 255) / 256, 256, 0, stream>>>(dst, scoreArr, smax_u, ssum);

  // Pass 4: weighted aggregation into out
  edge_agg_kernel<<<(N_EDGES / 16) / 4, 128, 0, stream>>>(
      src, dst, t, w_time, b_time, Wev, zv, scoreArr, ssum, out);
}